// Assessor_26225070309945
// MI455X (gfx1250) — compile-verified
//
#include <hip/hip_runtime.h>

// Problem geometry (fixed by the reference): (B=4, C=1, D=H=W=128) fp32.
#define BB 4
#define DD 128
#define HH 128
#define WW 128
#define N_TOTAL (BB * DD * HH * WW)   // 8,388,608
#define N_VEC   (N_TOTAL / 4)         // 2,097,152 float4 elements
#define SLICE   (HH * WW)             // 16,384

typedef float v2f __attribute__((ext_vector_type(2)));
typedef float v8f __attribute__((ext_vector_type(8)));

// ---------------------------------------------------------------------------
// Wave-level sum via V_WMMA_F32_16X16X4_F32 (the only f32 WMMA on CDNA5).
// A (16x4, 2 VGPRs): a.x = v, a.y = 0  ->  row m of A holds {v_m, 0, v_{m+16}, 0}.
// With B = ones(4x16): D[m][n] = v_m + v_{m+16}. Each lane's 8 C VGPRs cover
// half the wave's rows; one shfl_xor(16) completes the full 32-lane sum.
// Requires EXEC == all ones (callers use full blocks / uniform trip counts).
// ---------------------------------------------------------------------------
__device__ __forceinline__ float wave_sum_wmma(float v) {
    v2f a; a[0] = v;    a[1] = 0.0f;
    v2f b; b[0] = 1.0f; b[1] = 1.0f;
    v8f c = {};
    c = __builtin_amdgcn_wmma_f32_16x16x4_f32(
        /*neg_a=*/false, a, /*neg_b=*/false, b,
        /*c_mod=*/(short)0, c, /*reuse_a=*/false, /*reuse_b=*/false);
    float s = ((c[0] + c[1]) + (c[2] + c[3])) + ((c[4] + c[5]) + (c[6] + c[7]));
    s += __shfl_xor(s, 16, 32);
    return s;   // full 32-lane sum, uniform across the wave
}

// Block reduce (256 thr = 8 waves) + one atomic per block.
__device__ __forceinline__ void block_accum(float v, float* dst) {
    __shared__ float red[8];
    const float s = wave_sum_wmma(v);
    if ((threadIdx.x & 31) == 0) red[threadIdx.x >> 5] = s;
    __syncthreads();
    if (threadIdx.x == 0) {
        float t = 0.f;
        #pragma unroll
        for (int k = 0; k < 8; ++k) t += red[k];
        atomicAdd(dst, t);
    }
    __syncthreads();
}

// ---------------------------------------------------------------------------
// Branch-free masked loads: address is clamped to a safe in-bounds location,
// the loaded value is scaled by a 0/1 mask. Loads stay unconditional so the
// compiler can cluster them (no exec-mask diamonds, single wait for many).
// ---------------------------------------------------------------------------
__device__ __forceinline__ float4 ld4(const float* __restrict__ p, int i) {
    return *(const float4*)(p + i);
}
__device__ __forceinline__ float4 ld4m(const float* __restrict__ p, int i,
                                       int valid, int off) {
    const float4 v = *(const float4*)(p + i + (valid ? off : 0));
    const float  m = valid ? 1.0f : 0.0f;
    return make_float4(v.x * m, v.y * m, v.z * m, v.w * m);
}
__device__ __forceinline__ float ldm(const float* __restrict__ p, int i,
                                     int valid, int off) {
    return p[i + (valid ? off : 0)] * (valid ? 1.0f : 0.0f);
}

// 6-neighbor count for a float4 run along w (zero padding).
__device__ __forceinline__ float4 neigh6(const float* __restrict__ p, int i,
                                         int w0, int h, int d, float4 c) {
    const float  lft = ldm(p, i, w0 > 0,   -1);
    const float  rgt = ldm(p, i, w0 < 124,  4);
    const float4 up  = ld4m(p, i, h > 0,   -WW);
    const float4 dn  = ld4m(p, i, h < 127,  WW);
    const float4 bk  = ld4m(p, i, d > 0,   -SLICE);
    const float4 ft  = ld4m(p, i, d < 127,  SLICE);
    float4 nc;
    nc.x = lft + c.y + up.x + dn.x + bk.x + ft.x;
    nc.y = c.x + c.z + up.y + dn.y + bk.y + ft.y;
    nc.z = c.y + c.w + up.z + dn.z + bk.z + ft.z;
    nc.w = c.z + rgt + up.w + dn.w + bk.w + ft.w;
    return nc;
}

// ---------------------------------------------------------------------------
// Init: zero the 8 accumulators, write the flood-fill seed grid
// (m0[b,5,5,5] = 1 for every batch, 0 elsewhere).
// ---------------------------------------------------------------------------
__global__ void __launch_bounds__(256)
k_init(float* __restrict__ acc, float* __restrict__ m0) {
    const int stride = gridDim.x * blockDim.x;
    const int gid    = blockIdx.x * blockDim.x + threadIdx.x;
    if (gid < 8) acc[gid] = 0.0f;
    for (int t = gid; t < N_VEC; t += stride) {
        const int i = t * 4;
        const int w0 = i & 127, h = (i >> 7) & 127, d = (i >> 14) & 127;
        float4 v = make_float4(0.f, 0.f, 0.f, 0.f);
        if (d == 5 && h == 5 && w0 == 4) v.y = 1.0f;   // w = 5
        *(float4*)(m0 + i) = v;
    }
}

// ---------------------------------------------------------------------------
// Fused assessment pass: single sweep of x producing partial sums for
//   acc[0]=sum(x)  acc[1]=overhang  acc[2]=exposed  acc[3]=support  acc[4]=iso
// 4 voxels/thread, branch-free masked loads, grid-stride (uniform trips so
// EXEC stays all-1s for the WMMA reductions).
// ---------------------------------------------------------------------------
__global__ void __launch_bounds__(256)
k_assess(const float* __restrict__ x, float* __restrict__ acc) {
    float sum_x = 0.f, sum_ov = 0.f, sum_ex = 0.f, sum_sup = 0.f, sum_iso = 0.f;
    const int stride = gridDim.x * blockDim.x;            // in float4 units
    for (int t = blockIdx.x * blockDim.x + threadIdx.x; t < N_VEC; t += stride) {
        const int i  = t * 4;
        const int w0 = i & 127, h = (i >> 7) & 127, d = (i >> 14) & 127;

        // Pull the next stride's cacheline while we compute this one
        // (lowers to global_prefetch_b8; clamped address keeps it legal).
        const int pi = (t + stride < N_VEC) ? (t + stride) * 4 : i;
        __builtin_prefetch(x + pi, 0, 0);

        const float4 xc = ld4(x, i);
        const float4 nc = neigh6(x, i, w0, h, d, xc);

        sum_x += (xc.x + xc.y) + (xc.z + xc.w);

        sum_ex += fmaxf(6.f * xc.x - nc.x, 0.f) + fmaxf(6.f * xc.y - nc.y, 0.f)
                + fmaxf(6.f * xc.z - nc.z, 0.f) + fmaxf(6.f * xc.w - nc.w, 0.f);

        sum_sup += (nc.x <  2.f ? xc.x : 0.f) + (nc.y <  2.f ? xc.y : 0.f)
                 + (nc.z <  2.f ? xc.z : 0.f) + (nc.w <  2.f ? xc.w : 0.f);
        sum_iso += (nc.x <= 1.f ? xc.x : 0.f) + (nc.y <= 1.f ? xc.y : 0.f)
                 + (nc.z <= 1.f ? xc.z : 0.f) + (nc.w <= 1.f ? xc.w : 0.f);

        // Overhang: 3x3 mean of the slice below (d-1), zero-padded in H/W.
        const int dv = (d > 0);
        float4 s9 = make_float4(0.f, 0.f, 0.f, 0.f);
        #pragma unroll
        for (int dh = -1; dh <= 1; ++dh) {
            const int hv = dv && (h + dh >= 0) && (h + dh <= 127);
            const int ro = hv ? (dh * WW - SLICE) : 0;     // offset from i
            const float4 rv = ld4m(x, i, hv, ro);
            const float  rl = ldm(x, i, hv && (w0 > 0),   ro - 1);
            const float  rr = ldm(x, i, hv && (w0 < 124), ro + 4);
            s9.x += rl   + rv.x + rv.y;
            s9.y += rv.x + rv.y + rv.z;
            s9.z += rv.y + rv.z + rv.w;
            s9.w += rv.z + rv.w + rr;
        }
        const float dm = dv ? 1.0f : 0.0f;
        const float k9 = 1.0f / 9.0f;
        sum_ov += dm * (xc.x * (1.f - s9.x * k9) + xc.y * (1.f - s9.y * k9)
                      + xc.z * (1.f - s9.z * k9) + xc.w * (1.f - s9.w * k9));
    }

    block_accum(sum_x,   &acc[0]);
    block_accum(sum_ov,  &acc[1]);
    block_accum(sum_ex,  &acc[2]);
    block_accum(sum_sup, &acc[3]);
    block_accum(sum_iso, &acc[4]);
}

// ---------------------------------------------------------------------------
// One flood-fill iteration: dst = max(src, sigmoid(20*(nc(src)-0.5)))*(1-x).
// 4 voxels/thread, branch-free. x + both ping-pong buffers (96 MB) stay
// L2-resident (192 MB), so iterations 2..10 run out of L2.
// ---------------------------------------------------------------------------
__global__ void __launch_bounds__(256)
k_step(const float* __restrict__ x, const float* __restrict__ src,
       float* __restrict__ dst) {
    const int t = blockIdx.x * blockDim.x + threadIdx.x;
    if (t >= N_VEC) return;
    const int i  = t * 4;
    const int w0 = i & 127, h = (i >> 7) & 127, d = (i >> 14) & 127;

    const float4 c  = ld4(src, i);
    const float4 nc = neigh6(src, i, w0, h, d, c);
    const float4 xc = ld4(x, i);

    float4 r;
    r.x = fmaxf(c.x, 1.f / (1.f + __expf(-20.f * (nc.x - 0.5f)))) * (1.f - xc.x);
    r.y = fmaxf(c.y, 1.f / (1.f + __expf(-20.f * (nc.y - 0.5f)))) * (1.f - xc.y);
    r.z = fmaxf(c.z, 1.f / (1.f + __expf(-20.f * (nc.z - 0.5f)))) * (1.f - xc.z);
    r.w = fmaxf(c.w, 1.f / (1.f + __expf(-20.f * (nc.w - 0.5f)))) * (1.f - xc.w);
    *(float4*)(dst + i) = r;
}

// ---------------------------------------------------------------------------
// WMMA-backed grid reduction of an N_TOTAL array into acc[idx].
// 1024 blocks x 256 thr -> exactly 8 float4 per thread, uniform trip count.
// ---------------------------------------------------------------------------
__global__ void __launch_bounds__(256)
k_reduce(const float* __restrict__ v, float* __restrict__ acc, int idx) {
    float s = 0.f;
    const int stride = gridDim.x * blockDim.x;
    for (int t = blockIdx.x * blockDim.x + threadIdx.x; t < N_VEC; t += stride) {
        const float4 q = ld4(v, t * 4);
        s += (q.x + q.y) + (q.z + q.w);
    }
    block_accum(s, &acc[idx]);
}

// ---------------------------------------------------------------------------
// Finalize the 5 scalar outputs from the accumulators.
// ---------------------------------------------------------------------------
__global__ void k_final(const float* __restrict__ acc, float* __restrict__ out) {
    if (blockIdx.x == 0 && threadIdx.x == 0) {
        const float N   = (float)N_TOTAL;
        const float sx  = acc[0], ov = acc[1], ex = acc[2];
        const float sup = acc[3], iso = acc[4], vd = acc[5];
        const float docc = sx / N - 0.5f;
        out[0] = 10.0f * docc * docc;           // occupancy
        out[1] = ov / N;                        // overhang
        out[2] = ex / sx - 100.0f * vd / N;     // surface - trapped voids
        out[3] = sup / N;                       // support
        out[4] = iso / N;                       // isolation
    }
}

// ---------------------------------------------------------------------------
extern "C" void kernel_launch(void* const* d_in, const int* in_sizes, int n_in,
                              void* d_out, int out_size, void* d_ws, size_t ws_size,
                              hipStream_t stream) {
    (void)in_sizes; (void)n_in; (void)out_size; (void)ws_size;
    const float* x = (const float*)d_in[0];
    float* out     = (float*)d_out;

    float* acc = (float*)d_ws;            // 8 accumulator floats (16-f pad)
    float* m0  = acc + 16;                // flood-fill ping (N_TOTAL floats)
    float* m1  = m0 + N_TOTAL;            // flood-fill pong (N_TOTAL floats)

    k_init  <<<4096, 256, 0, stream>>>(acc, m0);
    k_assess<<<2048, 256, 0, stream>>>(x, acc);

    const int stepBlocks = N_VEC / 256;   // 8192
    float* src = m0;
    float* dst = m1;
    for (int it = 0; it < 10; ++it) {
        k_step<<<stepBlocks, 256, 0, stream>>>(x, src, dst);
        float* tmp = src; src = dst; dst = tmp;
    }
    // after 10 steps the result lives in `src` (== m0)
    k_reduce<<<1024, 256, 0, stream>>>(src, acc, 5);
    k_final <<<1, 32, 0, stream>>>(acc, out);
}